// LiquidConv50D2_25177098289493
// MI455X (gfx1250) — compile-verified
//
#include <hip/hip_runtime.h>
#include <hip/hip_bf16.h>

// ---------------- workspace layout (float offsets) ----------------
#define OFF_L1PACK   0          // 64*64 float4 = 16384 floats
#define OFF_L1SPACK  16384      // 32*64 float4 = 8192
#define OFF_L2PACK   24576      // 32*32 float4 = 4096
#define OFF_L2SPACK  28672      // 32*32 float4 = 4096
#define OFF_VEC1     32768      // 7 arrays of 64: cmt, gl, gv, bn, bd, sbn, sbd
#define OFF_VEC2     33216      // 7 arrays of 32
#define OFF_FCW      33440      // 16
#define OFF_FCB      33456      // 1
#define OFF_W1E      33472      // 8*32
#define OFF_B1E      33728      // 32
#define OFF_W2E      33760      // 32*32
#define OFF_B2E      34784      // 32
#define OFF_FEAT     36864      // N*32 floats
#define NROWS        (256*1024)

#define USE_BIG_LDS 1

// ---------------- fast tanh ----------------
#if defined(__AMDGCN__) && __has_builtin(__builtin_amdgcn_tanhf)
__device__ __forceinline__ float fast_tanh(float x) { return __builtin_amdgcn_tanhf(x); }
#else
__device__ __forceinline__ float fast_tanh(float x) {
  float e = __expf(-2.0f * x);
  return (1.0f - e) / (1.0f + e);
}
#endif

// ---------------- prep kernel args ----------------
struct PrepArgs {
  const float *c1w, *c1b, *c2w, *c2b;
  const float *g1, *vl1, *cm1, *sg1, *mu1, *w1, *er1;
  const float *ssg1, *smu1, *sw1, *ser1, *inw1, *inb1, *ow1, *ob1;
  const int   *mk1, *smk1;
  const float *g2, *vl2, *cm2, *sg2, *mu2, *w2, *er2;
  const float *ssg2, *smu2, *sw2, *ser2, *inw2, *inb2, *ow2, *ob2;
  const int   *mk2, *smk2;
  const float *fcw, *fcb;
};

__global__ void prep_kernel(PrepArgs A, float* __restrict__ ws) {
  const int tid = threadIdx.x;
  // Layer1 unfold pack: (sh, shm, wEh, wh) over (i=source 64, j=unit 64)
  for (int idx = tid; idx < 64 * 64; idx += 256) {
    float sh  = 0.5f * A.sg1[idx];
    float shm = sh * A.mu1[idx];
    float wm  = A.w1[idx] * (float)A.mk1[idx];
    float wh  = 0.5f * wm;
    float wEh = wh * A.er1[idx];
    ((float4*)(ws + OFF_L1PACK))[idx] = make_float4(sh, shm, wEh, wh);
  }
  // Layer1 synapse pack, fold input affine u = in_w*f + in_b
  for (int idx = tid; idx < 32 * 64; idx += 256) {
    int s = idx >> 6;
    float sg  = A.ssg1[idx];
    float sh  = 0.5f * sg * A.inw1[s];
    float shm = 0.5f * sg * (A.smu1[idx] - A.inb1[s]);
    float wm  = A.sw1[idx] * (float)A.smk1[idx];
    float wh  = 0.5f * wm;
    float wEh = wh * A.ser1[idx];
    ((float4*)(ws + OFF_L1SPACK))[idx] = make_float4(sh, shm, wEh, wh);
  }
  // Layer2 unfold pack
  for (int idx = tid; idx < 32 * 32; idx += 256) {
    float sh  = 0.5f * A.sg2[idx];
    float shm = sh * A.mu2[idx];
    float wm  = A.w2[idx] * (float)A.mk2[idx];
    float wh  = 0.5f * wm;
    float wEh = wh * A.er2[idx];
    ((float4*)(ws + OFF_L2PACK))[idx] = make_float4(sh, shm, wEh, wh);
  }
  // Layer2 synapse pack: input u = (v1*out_w1 + out_b1)*in_w2 + in_b2 folded
  for (int idx = tid; idx < 32 * 32; idx += 256) {
    int s = idx >> 5;
    float scale = A.ow1[s] * A.inw2[s];
    float bias  = A.ob1[s] * A.inw2[s] + A.inb2[s];
    float sg  = A.ssg2[idx];
    float sh  = 0.5f * sg * scale;
    float shm = 0.5f * sg * (A.smu2[idx] - bias);
    float wm  = A.sw2[idx] * (float)A.smk2[idx];
    float wh  = 0.5f * wm;
    float wEh = wh * A.ser2[idx];
    ((float4*)(ws + OFF_L2SPACK))[idx] = make_float4(sh, shm, wEh, wh);
  }
  // per-unit vectors, layer1
  if (tid < 64) {
    int j = tid;
    float bn = 0.f, bd = 0.f, sbn = 0.f, sbd = 0.f;
    for (int i = 0; i < 64; i++) {
      int k = i * 64 + j;
      float wm = A.w1[k] * (float)A.mk1[k];
      bd += 0.5f * wm; bn += 0.5f * wm * A.er1[k];
    }
    for (int s = 0; s < 32; s++) {
      int k = s * 64 + j;
      float wm = A.sw1[k] * (float)A.smk1[k];
      sbd += 0.5f * wm; sbn += 0.5f * wm * A.ser1[k];
    }
    ws[OFF_VEC1 + 0 * 64 + j] = A.cm1[j] * 6.0f;
    ws[OFF_VEC1 + 1 * 64 + j] = A.g1[j];
    ws[OFF_VEC1 + 2 * 64 + j] = A.g1[j] * A.vl1[j];
    ws[OFF_VEC1 + 3 * 64 + j] = bn;
    ws[OFF_VEC1 + 4 * 64 + j] = bd;
    ws[OFF_VEC1 + 5 * 64 + j] = sbn;
    ws[OFF_VEC1 + 6 * 64 + j] = sbd;
  }
  // per-unit vectors, layer2
  if (tid < 32) {
    int j = tid;
    float bn = 0.f, bd = 0.f, sbn = 0.f, sbd = 0.f;
    for (int i = 0; i < 32; i++) {
      int k = i * 32 + j;
      float wm = A.w2[k] * (float)A.mk2[k];
      bd += 0.5f * wm; bn += 0.5f * wm * A.er2[k];
    }
    for (int s = 0; s < 32; s++) {
      int k = s * 32 + j;
      float wm = A.sw2[k] * (float)A.smk2[k];
      sbd += 0.5f * wm; sbn += 0.5f * wm * A.ser2[k];
    }
    ws[OFF_VEC2 + 0 * 32 + j] = A.cm2[j] * 6.0f;
    ws[OFF_VEC2 + 1 * 32 + j] = A.g2[j];
    ws[OFF_VEC2 + 2 * 32 + j] = A.g2[j] * A.vl2[j];
    ws[OFF_VEC2 + 3 * 32 + j] = bn;
    ws[OFF_VEC2 + 4 * 32 + j] = bd;
    ws[OFF_VEC2 + 5 * 32 + j] = sbn;
    ws[OFF_VEC2 + 6 * 32 + j] = sbd;
  }
  // fc fold: logit = sum fc_w[m]*(v2[m]*ow2[m]+ob2[m]) + fc_b
  if (tid < 16) ws[OFF_FCW + tid] = A.fcw[tid] * A.ow2[tid];
  if (tid == 0) {
    float acc = A.fcb[0];
    for (int m = 0; m < 16; m++) acc += A.fcw[m] * A.ob2[m];
    ws[OFF_FCB] = acc;
  }
  // expanded conv weights, padded to 32 cols / 32 k
  for (int idx = tid; idx < 8 * 32; idx += 256) {
    int f = idx >> 5, col = idx & 31;
    float v = 0.f;
    if (col < 24) {
      int c = col / 6, wo = col % 6, k = f - wo;
      if (k >= 0 && k < 3) v = A.c1w[c * 3 + k];
    }
    ws[OFF_W1E + idx] = v;
  }
  if (tid < 32) ws[OFF_B1E + tid] = (tid < 24) ? A.c1b[tid / 6] : 0.f;
  for (int idx = tid; idx < 32 * 32; idx += 256) {
    int kin = idx >> 5, col = idx & 31;
    float v = 0.f;
    if (kin < 24) {
      int ci = kin / 6, wi = kin % 6;
      int c2 = col >> 2, wo2 = col & 3, k = wi - wo2;
      if (k >= 0 && k < 3) v = A.c2w[(c2 * 4 + ci) * 3 + k];
    }
    ws[OFF_W2E + idx] = v;
  }
  if (tid < 32) ws[OFF_B2E + tid] = A.c2b[tid >> 2];
}

// ---------------- conv front-end via WMMA f32 16x16x4 ----------------
typedef __attribute__((ext_vector_type(2))) float v2f;
typedef __attribute__((ext_vector_type(8))) float v8f;

__global__ __launch_bounds__(128) void conv_kernel(const float* __restrict__ x,
                                                   float* __restrict__ ws) {
#if defined(__AMDGCN__) && __has_builtin(__builtin_amdgcn_wmma_f32_16x16x4_f32)
  __shared__ float h1s[4][16][32];
  const int tid = threadIdx.x;
  const int ww = tid >> 5, l = tid & 31;
  const int r0 = (blockIdx.x * 4 + ww) * 16;
  const int m = l & 15, kh = l >> 4;
  const float* W1 = ws + OFF_W1E;
  const float* B1 = ws + OFF_B1E;
  const float* W2 = ws + OFF_W2E;
  const float* B2 = ws + OFF_B2E;
  float* feat = ws + OFF_FEAT;

  // Stage A: H1 = relu(X(16x8) * W1e(8x32) + b1)
  for (int nt = 0; nt < 2; nt++) {
    v8f acc = {};
    for (int kc = 0; kc < 2; kc++) {
      v2f a, bfr;
      const float* xp = x + (size_t)(r0 + m) * 8 + kc * 4 + kh * 2;
      a.x = xp[0]; a.y = xp[1];
      int krow = kc * 4 + kh * 2;
      bfr.x = W1[krow * 32 + nt * 16 + m];
      bfr.y = W1[(krow + 1) * 32 + nt * 16 + m];
      acc = __builtin_amdgcn_wmma_f32_16x16x4_f32(false, a, false, bfr,
                                                  (short)0, acc, false, false);
    }
    float bias = B1[nt * 16 + m];
    for (int vv = 0; vv < 8; vv++) {
      float h = acc[vv] + bias;
      h1s[ww][kh * 8 + vv][nt * 16 + m] = h > 0.f ? h : 0.f;
    }
  }
  __syncthreads();
  // Stage B: feat = H1(16x32) * W2e(32x32) + b2
  for (int nt = 0; nt < 2; nt++) {
    v8f acc = {};
    for (int kc = 0; kc < 8; kc++) {
      v2f a, bfr;
      int k0 = kc * 4 + kh * 2;
      a.x = h1s[ww][m][k0]; a.y = h1s[ww][m][k0 + 1];
      bfr.x = W2[k0 * 32 + nt * 16 + m];
      bfr.y = W2[(k0 + 1) * 32 + nt * 16 + m];
      acc = __builtin_amdgcn_wmma_f32_16x16x4_f32(false, a, false, bfr,
                                                  (short)0, acc, false, false);
    }
    float bias = B2[nt * 16 + m];
    for (int vv = 0; vv < 8; vv++)
      feat[(size_t)(r0 + kh * 8 + vv) * 32 + nt * 16 + m] = acc[vv] + bias;
  }
#else
  // scalar fallback (also used for the host pass parse)
  const int tid = threadIdx.x;
  const int base = blockIdx.x * 64;
  for (int e = tid; e < 64 * 32; e += 128) {
    int r = base + (e >> 5), col = e & 31;
    float acc = ws[OFF_B2E + col];
    for (int kin = 0; kin < 24; kin++) {
      float h = ws[OFF_B1E + kin];
      for (int f = 0; f < 8; f++)
        h = fmaf(x[(size_t)r * 8 + f], ws[OFF_W1E + f * 32 + kin], h);
      h = fmaxf(h, 0.f);
      acc = fmaf(h, ws[OFF_W2E + kin * 32 + col], acc);
    }
    ws[OFF_FEAT + (size_t)r * 32 + col] = acc;
  }
#endif
}

// ---------------- fused 2-layer LTC scan: one workgroup per batch ----------------
__global__ __launch_bounds__(256) void scan_kernel(const float* __restrict__ x,
                                                   float* __restrict__ ws,
                                                   float* __restrict__ out,
                                                   int useFeat) {
  const int b = blockIdx.x, tid = threadIdx.x;
  const int w = tid >> 5, l = tid & 31;

  __shared__ float v1[2][64], v2[2][32], u1[32];
  __shared__ float wns1[64], wds1[64], wns2[32], wds2[32];
  __shared__ float h1tmp[24], xrow[8];
  __shared__ float cmt1[64], gl1[64], gv1[64], bn1[64], bd1[64], sbn1[64], sbd1[64];
  __shared__ float cmt2[32], gl2[32], gv2[32], bn2[32], bd2[32], sbn2[32], sbd2[32];

#if USE_BIG_LDS
  __shared__ float4 sl1p[64 * 64];   // 64 KB
  __shared__ float4 sl1sp[32 * 64];  // 32 KB
  __shared__ float4 sl2p[32 * 32];   // 16 KB
  __shared__ float4 sl2sp[32 * 32];  // 16 KB
  for (int i = tid; i < 4096; i += 256) sl1p[i]  = ((const float4*)(ws + OFF_L1PACK))[i];
  for (int i = tid; i < 2048; i += 256) sl1sp[i] = ((const float4*)(ws + OFF_L1SPACK))[i];
  for (int i = tid; i < 1024; i += 256) sl2p[i]  = ((const float4*)(ws + OFF_L2PACK))[i];
  for (int i = tid; i < 1024; i += 256) sl2sp[i] = ((const float4*)(ws + OFF_L2SPACK))[i];
  #define L1P(k)  sl1p[k]
  #define L1SP(k) sl1sp[k]
  #define L2P(k)  sl2p[k]
  #define L2SP(k) sl2sp[k]
#else
  const float4* g_l1p  = (const float4*)(ws + OFF_L1PACK);
  const float4* g_l1sp = (const float4*)(ws + OFF_L1SPACK);
  const float4* g_l2p  = (const float4*)(ws + OFF_L2PACK);
  const float4* g_l2sp = (const float4*)(ws + OFF_L2SPACK);
  #define L1P(k)  g_l1p[k]
  #define L1SP(k) g_l1sp[k]
  #define L2P(k)  g_l2p[k]
  #define L2SP(k) g_l2sp[k]
#endif

  if (tid < 64) {
    cmt1[tid] = ws[OFF_VEC1 + tid];
    gl1[tid]  = ws[OFF_VEC1 + 64 + tid];
    gv1[tid]  = ws[OFF_VEC1 + 128 + tid];
    bn1[tid]  = ws[OFF_VEC1 + 192 + tid];
    bd1[tid]  = ws[OFF_VEC1 + 256 + tid];
    sbn1[tid] = ws[OFF_VEC1 + 320 + tid];
    sbd1[tid] = ws[OFF_VEC1 + 384 + tid];
    v1[0][tid] = 0.f; v1[1][tid] = 0.f;
  }
  if (tid < 32) {
    cmt2[tid] = ws[OFF_VEC2 + tid];
    gl2[tid]  = ws[OFF_VEC2 + 32 + tid];
    gv2[tid]  = ws[OFF_VEC2 + 64 + tid];
    bn2[tid]  = ws[OFF_VEC2 + 96 + tid];
    bd2[tid]  = ws[OFF_VEC2 + 128 + tid];
    sbn2[tid] = ws[OFF_VEC2 + 160 + tid];
    sbd2[tid] = ws[OFF_VEC2 + 192 + tid];
    v2[0][tid] = 0.f; v2[1][tid] = 0.f;
  }
  __syncthreads();

  const int j1 = (w << 3) | (l & 7), ig1 = l >> 3;  // 64 units x 4 src-chunks (in-wave)
  const int j2 = (w << 2) | (l & 3), ig2 = l >> 2;  // 32 units x 8 src-chunks (in-wave)
  const float* feat = ws + OFF_FEAT + (size_t)b * 1024 * 32;
  const float* xr = x + (size_t)b * 1024 * 8;
  int c1 = 0, c2 = 0;

  for (int t = 0; t < 1024; t++) {
    // ---- features for this timestep ----
    if (useFeat) {
      if (tid < 32) u1[tid] = feat[t * 32 + tid];
    } else {
      if (tid < 8) xrow[tid] = xr[t * 8 + tid];
      __syncthreads();
      if (tid < 24) {
        float s = ws[OFF_B1E + tid];
        for (int f = 0; f < 8; f++) s = fmaf(xrow[f], ws[OFF_W1E + f * 32 + tid], s);
        h1tmp[tid] = fmaxf(s, 0.f);
      }
      __syncthreads();
      if (tid < 32) {
        float s = ws[OFF_B2E + tid];
        for (int k = 0; k < 24; k++) s = fmaf(h1tmp[k], ws[OFF_W2E + k * 32 + tid], s);
        u1[tid] = s;
      }
    }
    __syncthreads();

    // ---- layer1 synapse (per timestep) ----
    {
      float an = 0.f, ad = 0.f;
#pragma unroll
      for (int ss = 0; ss < 8; ss++) {
        int s = ig1 * 8 + ss;
        float4 p = L1SP(s * 64 + j1);
        float tt = fast_tanh(fmaf(p.x, u1[s], -p.y));
        an = fmaf(p.z, tt, an); ad = fmaf(p.w, tt, ad);
      }
      an += __shfl_xor(an, 8);  an += __shfl_xor(an, 16);
      ad += __shfl_xor(ad, 8);  ad += __shfl_xor(ad, 16);
      if (ig1 == 0) { wns1[j1] = sbn1[j1] + an; wds1[j1] = sbd1[j1] + ad; }
    }
    __syncthreads();

    // ---- layer1 unfolds ----
    for (int uu = 0; uu < 6; uu++) {
      float an = 0.f, ad = 0.f;
#pragma unroll
      for (int ii = 0; ii < 16; ii++) {
        int i = ig1 * 16 + ii;
        float4 p = L1P(i * 64 + j1);
        float tt = fast_tanh(fmaf(p.x, v1[c1][i], -p.y));
        an = fmaf(p.z, tt, an); ad = fmaf(p.w, tt, ad);
      }
      an += __shfl_xor(an, 8);  an += __shfl_xor(an, 16);
      ad += __shfl_xor(ad, 8);  ad += __shfl_xor(ad, 16);
      if (ig1 == 0) {
        float vv = v1[c1][j1];
        float num = fmaf(cmt1[j1], vv, gv1[j1]) + bn1[j1] + an + wns1[j1];
        float den = cmt1[j1] + gl1[j1] + bd1[j1] + ad + wds1[j1] + 1e-8f;
        v1[c1 ^ 1][j1] = num / den;
      }
      __syncthreads();
      c1 ^= 1;
    }

    // ---- layer2 synapse (input = layer1 output, affine folded) ----
    {
      float an = 0.f, ad = 0.f;
#pragma unroll
      for (int ss = 0; ss < 4; ss++) {
        int s = ig2 * 4 + ss;
        float4 p = L2SP(s * 32 + j2);
        float tt = fast_tanh(fmaf(p.x, v1[c1][s], -p.y));
        an = fmaf(p.z, tt, an); ad = fmaf(p.w, tt, ad);
      }
      an += __shfl_xor(an, 4); an += __shfl_xor(an, 8); an += __shfl_xor(an, 16);
      ad += __shfl_xor(ad, 4); ad += __shfl_xor(ad, 8); ad += __shfl_xor(ad, 16);
      if (ig2 == 0) { wns2[j2] = sbn2[j2] + an; wds2[j2] = sbd2[j2] + ad; }
    }
    __syncthreads();

    // ---- layer2 unfolds ----
    for (int uu = 0; uu < 6; uu++) {
      float an = 0.f, ad = 0.f;
#pragma unroll
      for (int ii = 0; ii < 4; ii++) {
        int i = ig2 * 4 + ii;
        float4 p = L2P(i * 32 + j2);
        float tt = fast_tanh(fmaf(p.x, v2[c2][i], -p.y));
        an = fmaf(p.z, tt, an); ad = fmaf(p.w, tt, ad);
      }
      an += __shfl_xor(an, 4); an += __shfl_xor(an, 8); an += __shfl_xor(an, 16);
      ad += __shfl_xor(ad, 4); ad += __shfl_xor(ad, 8); ad += __shfl_xor(ad, 16);
      if (ig2 == 0) {
        float vv = v2[c2][j2];
        float num = fmaf(cmt2[j2], vv, gv2[j2]) + bn2[j2] + an + wns2[j2];
        float den = cmt2[j2] + gl2[j2] + bd2[j2] + ad + wds2[j2] + 1e-8f;
        v2[c2 ^ 1][j2] = num / den;
      }
      __syncthreads();
      c2 ^= 1;
    }
  }

  if (tid == 0) {
    float acc = ws[OFF_FCB];
    for (int m = 0; m < 16; m++) acc = fmaf(ws[OFF_FCW + m], v2[c2][m], acc);
    out[b] = 1.0f / (1.0f + __expf(-acc));
  }
}

// ---------------- launch ----------------
extern "C" void kernel_launch(void* const* d_in, const int* in_sizes, int n_in,
                              void* d_out, int out_size, void* d_ws, size_t ws_size,
                              hipStream_t stream) {
  PrepArgs A;
  A.c1w  = (const float*)d_in[1];  A.c1b  = (const float*)d_in[2];
  A.c2w  = (const float*)d_in[3];  A.c2b  = (const float*)d_in[4];
  A.g1   = (const float*)d_in[5];  A.vl1  = (const float*)d_in[6];
  A.cm1  = (const float*)d_in[7];  A.sg1  = (const float*)d_in[8];
  A.mu1  = (const float*)d_in[9];  A.w1   = (const float*)d_in[10];
  A.er1  = (const float*)d_in[11]; A.ssg1 = (const float*)d_in[12];
  A.smu1 = (const float*)d_in[13]; A.sw1  = (const float*)d_in[14];
  A.ser1 = (const float*)d_in[15]; A.inw1 = (const float*)d_in[16];
  A.inb1 = (const float*)d_in[17]; A.ow1  = (const float*)d_in[18];
  A.ob1  = (const float*)d_in[19];
  A.mk1  = (const int*)d_in[20];   A.smk1 = (const int*)d_in[21];
  A.g2   = (const float*)d_in[22]; A.vl2  = (const float*)d_in[23];
  A.cm2  = (const float*)d_in[24]; A.sg2  = (const float*)d_in[25];
  A.mu2  = (const float*)d_in[26]; A.w2   = (const float*)d_in[27];
  A.er2  = (const float*)d_in[28]; A.ssg2 = (const float*)d_in[29];
  A.smu2 = (const float*)d_in[30]; A.sw2  = (const float*)d_in[31];
  A.ser2 = (const float*)d_in[32]; A.inw2 = (const float*)d_in[33];
  A.inb2 = (const float*)d_in[34]; A.ow2  = (const float*)d_in[35];
  A.ob2  = (const float*)d_in[36];
  A.mk2  = (const int*)d_in[37];   A.smk2 = (const int*)d_in[38];
  A.fcw  = (const float*)d_in[39]; A.fcb  = (const float*)d_in[40];

  float* ws = (float*)d_ws;
  const float* x = (const float*)d_in[0];
  float* out = (float*)d_out;

  prep_kernel<<<1, 256, 0, stream>>>(A, ws);

  size_t need = (size_t)(OFF_FEAT + (size_t)NROWS * 32) * sizeof(float);
  int useFeat = (ws_size >= need) ? 1 : 0;
  if (useFeat) conv_kernel<<<NROWS / 64, 128, 0, stream>>>(x, ws);

  scan_kernel<<<256, 256, 0, stream>>>(x, ws, out, useFeat);
}